// Gcn_37976100831416
// MI455X (gfx1250) — compile-verified
//
#include <hip/hip_runtime.h>
#include <hip/hip_bf16.h>

typedef __attribute__((ext_vector_type(2))) float v2f;
typedef __attribute__((ext_vector_type(8))) float v8f;

#define IN_F 256
#define OUT_F 128

// -------------------------------------------------------------------------
// Kernel 1: support = x @ W using V_WMMA_F32_16X16X4_F32.
// One wave owns one 16-row M-tile across the FULL N=128 (8 C accumulators).
// The streaming operand (x, 102 MB) is read exactly once; the tiny shared
// operand (W, 128 KB) is re-read from L1/L2 where it is resident.
// A 16x4 f32 fragment (ISA 7.12.2): lanes 0-15 -> M=lane, K={0,1};
// lanes 16-31 -> M=lane-16, K={2,3}. B 4x16 mirrors with N across lanes.
// -------------------------------------------------------------------------
__global__ void gcn_gemm_wmma(const float* __restrict__ x,
                              const float* __restrict__ w,
                              float* __restrict__ support,
                              int nNodes) {
    const int lane  = threadIdx.x & 31;
    const int waveG = (blockIdx.x * blockDim.x + threadIdx.x) >> 5; // M tile
    const int m0    = waveG * 16;
    if (m0 >= nNodes) return;               // wave-uniform: EXEC stays all-1s

    const int mrow  = lane & 15;
    const int ncol  = lane & 15;
    const int khalf = (lane >> 4) * 2;      // 0 for lanes 0-15, 2 for 16-31

    const float* arow = x + (size_t)(m0 + mrow) * IN_F + khalf;
    const float* brow = w + (size_t)khalf * OUT_F + ncol;

    v8f c[8] = {};
    for (int k = 0; k < IN_F; k += 4) {
        v2f a = *(const v2f*)arow;          // K = khalf, khalf+1 (8B aligned)
        arow += 4;
        #pragma unroll
        for (int n = 0; n < 8; ++n) {
            v2f b;
            b.x = brow[n * 16];             // row khalf,   col 16n + ncol
            b.y = brow[n * 16 + OUT_F];     // row khalf+1, col 16n + ncol
            c[n] = __builtin_amdgcn_wmma_f32_16x16x4_f32(
                       /*neg_a=*/false, a, /*neg_b=*/false, b,
                       /*c_mod=*/(short)0, c[n],
                       /*reuse_a=*/false, /*reuse_b=*/false);
        }
        brow += 4 * OUT_F;
    }

    // C/D layout: VGPR j, lanes 0-15 -> M=j, lanes 16-31 -> M=j+8; N=lane&15
    float* dst = support + (size_t)(m0 + (lane >> 4) * 8) * OUT_F + ncol;
    #pragma unroll
    for (int n = 0; n < 8; ++n) {
        #pragma unroll
        for (int j = 0; j < 8; ++j) {
            dst[(size_t)j * OUT_F + n * 16] = c[n][j];
        }
    }
}

// -------------------------------------------------------------------------
// Kernel 2: out[n][f] = bias[f]  (float4 vectorized; d_out is poisoned)
// -------------------------------------------------------------------------
__global__ void gcn_bias_init(float* __restrict__ out,
                              const float* __restrict__ bias,
                              int total4) {
    int i = blockIdx.x * blockDim.x + threadIdx.x;
    if (i >= total4) return;
    const float4* b4 = (const float4*)bias;
    ((float4*)out)[i] = b4[i & (OUT_F / 4 - 1)];
}

// -------------------------------------------------------------------------
// Kernel 3: for each edge e: out[row[e]] += val[e] * support[col[e]]
// One wave per edge: lane handles 4 features (float4 gather, 4 f32 atomics).
// support (51 MB) and out (51 MB) both fit in the 192 MB L2, so the random
// gathers and the scatter atomics stay L2-resident.
// -------------------------------------------------------------------------
__global__ void gcn_edge_scatter(const float* __restrict__ support,
                                 const int* __restrict__ erow,
                                 const int* __restrict__ ecol,
                                 const float* __restrict__ eval,
                                 float* __restrict__ out,
                                 int nEdges) {
    const int edgesPerBlock = blockDim.x >> 5;
    const int e = blockIdx.x * edgesPerBlock + (threadIdx.x >> 5);
    if (e >= nEdges) return;
    const int lane = threadIdx.x & 31;

    const int   r = erow[e];
    const int   c = ecol[e];
    const float v = eval[e];

    float4 g = ((const float4*)(support + (size_t)c * OUT_F))[lane];
    float* dst = out + (size_t)r * OUT_F + lane * 4;
    atomicAdd(dst + 0, v * g.x);
    atomicAdd(dst + 1, v * g.y);
    atomicAdd(dst + 2, v * g.z);
    atomicAdd(dst + 3, v * g.w);
}

extern "C" void kernel_launch(void* const* d_in, const int* in_sizes, int n_in,
                              void* d_out, int out_size, void* d_ws, size_t ws_size,
                              hipStream_t stream) {
    const float* x    = (const float*)d_in[0];
    const int*   erow = (const int*)  d_in[1];
    const int*   ecol = (const int*)  d_in[2];
    const float* ev   = (const float*)d_in[3];
    const float* w    = (const float*)d_in[4];
    const float* bias = (const float*)d_in[5];
    float* out     = (float*)d_out;
    float* support = (float*)d_ws;   // N_NODES * OUT_F floats = 51.2 MB

    const int nNodes = in_sizes[0] / IN_F;   // 100000 (divisible by 16)
    const int nEdges = in_sizes[1];          // 1600000

    // 1) support = x @ W  (fp32 WMMA, 1 wave per 16-row M-tile, full N)
    const int mTiles = nNodes / 16;                       // 6250 waves
    const int wavesPerBlock = 256 / 32;                   // 8
    const int gemmBlocks = (mTiles + wavesPerBlock - 1) / wavesPerBlock;
    gcn_gemm_wmma<<<gemmBlocks, 256, 0, stream>>>(x, w, support, nNodes);

    // 2) out = bias (broadcast)
    const int total4 = nNodes * (OUT_F / 4);
    gcn_bias_init<<<(total4 + 255) / 256, 256, 0, stream>>>(out, bias, total4);

    // 3) out[row] += val * support[col]  (atomic scatter, L2-resident)
    const int edgesPerBlock = 256 / 32;
    gcn_edge_scatter<<<(nEdges + edgesPerBlock - 1) / edgesPerBlock, 256, 0, stream>>>(
        support, erow, ecol, ev, out, nEdges);
}